// LabelSmoothingLoss_61667140436165
// MI455X (gfx1250) — compile-verified
//
#include <hip/hip_runtime.h>
#include <stdint.h>

// LabelSmoothingLoss, fused single pass for gfx1250 (MI455X).
//   loss[b] = -sum_c smooth(marker[c]) * log_sigmoid(output[b,c])
// output[] is streamed through LDS with CDNA5 async global->LDS copies
// (global_load_async_to_lds_b128, ASYNCcnt), 4-deep software pipeline.

#define NT      32000
#define LSEQ    50
#define BLOCK   320                  // 10 wave32's
#define TILE    (BLOCK * 4)          // 1280 floats per stage (5120 B)
#define NSTAGES (NT / TILE)          // 25, exact
#define DEPTH   4
static_assert(NT % TILE == 0, "tile must divide NT exactly");
static_assert(NSTAGES > DEPTH, "pipeline deeper than stage count");

__device__ __forceinline__ float logsig(float x) {
    // log_sigmoid(x) = min(x,0) - log(1 + exp(-|x|)) ; v_exp_f32 / v_log_f32
    return fminf(x, 0.0f) - __logf(1.0f + __expf(-fabsf(x)));
}

__launch_bounds__(BLOCK)
__global__ void lsl_kernel(const float* __restrict__ out,
                           const int*   __restrict__ label,
                           const int*   __restrict__ test_label,
                           float*       __restrict__ loss) {
    __shared__ uint32_t marker32[NT / 4];               // 32 KB: 1 byte/class
    __shared__ __align__(16) float stage[DEPTH][TILE];  // 20 KB: async landing
    __shared__ float wsum[BLOCK / 32];
    uint8_t* marker = (uint8_t*)marker32;

    const int b   = blockIdx.x;
    const int tid = threadIdx.x;

    const float* row  = out + (size_t)b * NT;            // uniform -> SGPR pair
    const uint32_t goff    = (uint32_t)tid * 16u;        // lane byte off in row
    const uint32_t ldsbase = (uint32_t)(uintptr_t)&stage[0][0] + (uint32_t)tid * 16u;

    // ---- prime DEPTH async stages immediately: overlap HBM latency with
    //      marker construction below ----
#pragma unroll
    for (int p = 0; p < DEPTH; ++p) {
        const uint32_t la = ldsbase + (uint32_t)(p * TILE * 4);
        const uint32_t ga = goff + (uint32_t)(p * TILE * 4);
        asm volatile("global_load_async_to_lds_b128 %0, %1, %2"
                     :: "v"(la), "v"(ga), "s"(row) : "memory");
    }

    // ---- build per-row class markers in LDS (set semantics + overwrite) ----
    {
        uint4* m128 = (uint4*)marker32;                  // 2000 x 16B
        const uint4 z = {0u, 0u, 0u, 0u};
        for (int i = tid; i < NT / 16; i += BLOCK) m128[i] = z;
    }
    __syncthreads();
    if (tid < LSEQ) {
        int v = label[b * LSEQ + tid];
        if (v > 0) marker[v - 1] = (uint8_t)1;           // same-value races benign
    }
    __syncthreads();
    if (tid < LSEQ) {
        int v = test_label[b * LSEQ + tid];
        if (v > 0) marker[v - 1] = (uint8_t)2;           // test overwrites label
    }
    __syncthreads();

    const float base = 0.1f / (float)NT;                 // smooth(0) = eps/N
    const float c2   = 1.8f - base;                      // smooth(2)

    float acc = 0.0f;

    // each lane reads back exactly the 16B it staged: no barrier needed,
    // only the per-wave ASYNCcnt wait.
    auto consume = [&](int s) {
        const float4 x = *(const float4*)&stage[s & (DEPTH - 1)][tid * 4];
        const uint32_t m = marker32[s * (TILE / 4) + tid]; // 4 marker bytes
        // drain DS before this LDS buffer can be re-targeted by a new async op
        asm volatile("s_wait_dscnt 0" ::: "memory");
        const float xs[4] = {x.x, x.y, x.z, x.w};
#pragma unroll
        for (int j = 0; j < 4; ++j) {
            const uint32_t v = (m >> (8 * j)) & 0xffu;
            const float coef = (v == 1u) ? 0.9f : ((v == 2u) ? c2 : base);
            acc = fmaf(coef, logsig(xs[j]), acc);
        }
    };

    // ---- main pipeline: wait oldest-of-4, consume, refill ----
    for (int s = 0; s < NSTAGES - DEPTH; ++s) {
        asm volatile("s_wait_asynccnt 3" ::: "memory");  // stage s complete
        consume(s);
        const uint32_t la = ldsbase + (uint32_t)((s & (DEPTH - 1)) * TILE * 4);
        const uint32_t ga = goff + (uint32_t)((s + DEPTH) * TILE * 4);
        asm volatile("global_load_async_to_lds_b128 %0, %1, %2"
                     :: "v"(la), "v"(ga), "s"(row) : "memory");
    }
    // ---- epilogue: drain the last DEPTH stages ----
    asm volatile("s_wait_asynccnt 3" ::: "memory"); consume(NSTAGES - 4);
    asm volatile("s_wait_asynccnt 2" ::: "memory"); consume(NSTAGES - 3);
    asm volatile("s_wait_asynccnt 1" ::: "memory"); consume(NSTAGES - 2);
    asm volatile("s_wait_asynccnt 0" ::: "memory"); consume(NSTAGES - 1);

    // ---- reduction: shfl within wave32, then across the 10 waves via LDS ----
#pragma unroll
    for (int off = 16; off > 0; off >>= 1)
        acc += __shfl_down(acc, off, 32);
    if ((tid & 31) == 0) wsum[tid >> 5] = acc;
    __syncthreads();
    if (tid == 0) {
        float t = 0.0f;
#pragma unroll
        for (int w = 0; w < BLOCK / 32; ++w) t += wsum[w];
        loss[b] = -t;
    }
}

extern "C" void kernel_launch(void* const* d_in, const int* in_sizes, int n_in,
                              void* d_out, int out_size, void* d_ws, size_t ws_size,
                              hipStream_t stream) {
    const float* output     = (const float*)d_in[0];   // (B, NT) f32
    const int*   label      = (const int*)d_in[1];     // (B, L)  i32
    const int*   test_label = (const int*)d_in[2];     // (B, L)  i32
    // d_in[3] = num_type scalar (32000), compile-time constant here

    const int B = in_sizes[1] / LSEQ;                  // 4096
    float* loss = (float*)d_out;                       // (B,) f32

    lsl_kernel<<<B, BLOCK, 0, stream>>>(output, label, test_label, loss);
}